// TransformerSwapGNN_4604204941421
// MI455X (gfx1250) — compile-verified
//
#include <hip/hip_runtime.h>
#include <hip/hip_bf16.h>
#include <stdint.h>

// ---------------------------------------------------------------------------
// TransformerSwapGNN for MI455X (gfx1250).  L2-resident, bandwidth/latency
// bound workload.  Dense GEMMs use V_WMMA_F32_16X16X4_F32 with branch-free
// inner loops (row-clamped A b64 loads, transposed LDS-staged B -> single
// aligned ds_load_b64 per fragment); W staged via gfx1250 async global->LDS.
// Edge attention via global float atomics; BN stats via LDS atomics.
// ---------------------------------------------------------------------------

typedef float v2f __attribute__((ext_vector_type(2)));
typedef float v8f __attribute__((ext_vector_type(8)));

#define HAVE_WMMA_F32 __has_builtin(__builtin_amdgcn_wmma_f32_16x16x4_f32)
#define HAVE_ASYNC_LDS __has_builtin(__builtin_amdgcn_global_load_async_to_lds_b32)

#define NEGBIG (-1e9f)

// ---------------- ordered-int encoding for float atomic max ----------------
__device__ __host__ inline unsigned fkey(float f) {
  unsigned u;
  __builtin_memcpy(&u, &f, 4);
  return (u & 0x80000000u) ? ~u : (u | 0x80000000u);
}
__device__ inline float fdec(unsigned k) {
  unsigned u = (k & 0x80000000u) ? (k & 0x7FFFFFFFu) : ~k;
  float f;
  __builtin_memcpy(&f, &u, 4);
  return f;
}

// ---------------- threefry2x32 (JAX PRNG), host+device ---------------------
__host__ __device__ inline void threefry2x32(uint32_t k0, uint32_t k1,
                                             uint32_t c0, uint32_t c1,
                                             uint32_t& o0, uint32_t& o1) {
  const uint32_t rot[8] = {13, 15, 26, 6, 17, 29, 16, 24};
  uint32_t ks[3] = {k0, k1, k0 ^ k1 ^ 0x1BD11BDAu};
  uint32_t x0 = c0 + ks[0], x1 = c1 + ks[1];
  for (int g = 0; g < 5; ++g) {
    for (int r = 0; r < 4; ++r) {
      uint32_t R = rot[(g & 1) * 4 + r];
      x0 += x1;
      x1 = (x1 << R) | (x1 >> (32 - R));
      x1 ^= x0;
    }
    x0 += ks[(g + 1) % 3];
    x1 += ks[(g + 2) % 3] + (uint32_t)(g + 1);
  }
  o0 = x0;
  o1 = x1;
}

__device__ inline float gumbel_of(uint32_t k0, uint32_t k1, uint32_t n) {
  uint32_t o0, o1;
  threefry2x32(k0, k1, n, 0u, o0, o1);
  unsigned m = 0x3F800000u | (o0 >> 9);
  float u;
  __builtin_memcpy(&u, &m, 4);
  u -= 1.0f;
  u = fmaxf(u, 1e-12f);
  return -__logf(-__logf(u));
}

// ---------------------------------------------------------------------------
// WMMA GEMM, K/M compile-time (K % 4 == 0, M % 16 == 0, M <= 128).
// Block = 256 threads = 8 waves; wave w covers rows [ (blk*8+w)*16, +16 ) and
// ALL M columns (M/16 accumulators): A fragment loaded once per K-step.
// W is staged TRANSPOSED in LDS (sWt[col*(K+2)+k], pad keeps b64 alignment
// and breaks pow2 bank stride) so each B fragment is ONE aligned ds_load_b64
// into an even register pair -> no repacking movs before the WMMA.
// Staging uses gfx1250 async global->LDS (per-lane src/dst pointers allow the
// transposing scatter with coalesced global reads).
// Fragment layouts per CDNA5 ISA 7.12.2:
//   A 16x4 : lane L: row = L&15, VGPRs = K{0,1} (lanes 0-15) / K{2,3}
//   B 4x16 : lane L: col = L&15, VGPRs = K{0,1} (lanes 0-15) / K{2,3}
//   C/D    : VGPR j, lane L: row = j + 8*(L>>4), col = L&15
// Out-of-range rows are CLAMPED on load (their D rows are simply not stored),
// keeping the inner loop branch-free and EXEC all-ones.
// ---------------------------------------------------------------------------
template <int K, int M>
__global__ __launch_bounds__(256) void gemm_wmma(
    const float* __restrict__ X, const float* __restrict__ W,
    const float* __restrict__ bias, float* __restrict__ Y, int N, int relu) {
  constexpr int KP = K + 2;  // padded col stride (even -> b64-aligned pairs)
  __shared__ float sWt[M * KP];

  // ---- stage W transposed into LDS (whole block cooperates) ----
#if HAVE_ASYNC_LDS
  {
    typedef __attribute__((address_space(1))) int gint;
    typedef __attribute__((address_space(3))) int lint;
    for (int i = threadIdx.x; i < K * M; i += 256) {
      int k = i / M, col = i % M;  // coalesced global read, scattered LDS write
      __builtin_amdgcn_global_load_async_to_lds_b32(
          (gint*)(W + i), (lint*)(sWt + col * KP + k), 0, 0);
    }
#if __has_builtin(__builtin_amdgcn_s_wait_asynccnt)
    __builtin_amdgcn_s_wait_asynccnt(0);
#else
    asm volatile("s_wait_asynccnt 0x0" ::: "memory");
#endif
  }
#else
  for (int i = threadIdx.x; i < K * M; i += 256) {
    int k = i / M, col = i % M;
    sWt[col * KP + k] = W[i];
  }
#endif
  __syncthreads();

  int lane = threadIdx.x & 31;
  int wave = threadIdx.x >> 5;
  int rowTile = blockIdx.x * 8 + wave;
  if (rowTile >= (N + 15) >> 4) return;  // uniform per wave

  int half = lane >> 4;
  int mr = lane & 15;
  int row = rowTile * 16 + mr;
  int rclamp = row < N ? row : N - 1;
  const float* xrow = X + (long)rclamp * K;

  constexpr int MT = M / 16;
  v8f acc[MT];
#pragma unroll
  for (int mt = 0; mt < MT; ++mt) acc[mt] = (v8f){};

#if HAVE_WMMA_F32
  for (int k0 = 0; k0 < K; k0 += 4) {
    int ka = k0 + half * 2;
    v2f a = *(const v2f*)(xrow + ka);  // one global b64 load, 8B aligned
#pragma unroll
    for (int mt = 0; mt < MT; ++mt) {
      // one aligned ds_load_b64: (ka, ka+1) contiguous in transposed layout
      v2f b = *(const v2f*)(sWt + (mt * 16 + mr) * KP + ka);
      acc[mt] = __builtin_amdgcn_wmma_f32_16x16x4_f32(false, a, false, b,
                                                      (short)0, acc[mt],
                                                      false, false);
    }
  }
#else
  for (int k = 0; k < K; ++k) {
#pragma unroll
    for (int mt = 0; mt < MT; ++mt) {
      float wv = sWt[(mt * 16 + mr) * KP + k];
#pragma unroll
      for (int j = 0; j < 8; ++j) {
        int r = rowTile * 16 + j + half * 8;
        int rc = r < N ? r : N - 1;
        acc[mt][j] += X[(long)rc * K + k] * wv;
      }
    }
  }
#endif

#pragma unroll
  for (int mt = 0; mt < MT; ++mt) {
    float bv = bias[mt * 16 + mr];
#pragma unroll
    for (int j = 0; j < 8; ++j) {
      int r = rowTile * 16 + j + half * 8;
      if (r < N) {
        float y = acc[mt][j] + bv;
        if (relu) y = fmaxf(y, 0.f);
        Y[(long)r * M + mt * 16 + mr] = y;
      }
    }
  }
}

// ---------------- first layer GEMM (K=2 outer product) ----------------------
__global__ void gemm_k2_kernel(const float* __restrict__ X2,
                               const float* __restrict__ W,
                               const float* __restrict__ b,
                               float* __restrict__ Y, int N) {
  long id = (long)blockIdx.x * blockDim.x + threadIdx.x;
  if (id >= (long)N * 128) return;
  int n = (int)(id >> 7), m = (int)(id & 127);
  Y[id] = X2[(long)n * 2] * W[m] + X2[(long)n * 2 + 1] * W[128 + m] + b[m];
}

// ---------------- stack([node_type, requests], axis=1) ----------------------
__global__ void stack_kernel(const float* nt, const float* rq, float* X2, int N) {
  int n = blockIdx.x * blockDim.x + threadIdx.x;
  if (n >= N) return;
  X2[(long)n * 2 + 0] = nt[n];
  X2[(long)n * 2 + 1] = rq[n];
}

// ---------------- edge attention: alpha + segment max -----------------------
__global__ __launch_bounds__(128) void edge_alpha_kernel(
    const float* __restrict__ q, const float* __restrict__ k,
    const float* __restrict__ We, const float* __restrict__ lat,
    const int* __restrict__ ei, float* __restrict__ alpha,
    unsigned* __restrict__ amax, int E) {
  int e = blockIdx.x;
  int c = threadIdx.x;  // 0..127 ; wave = one head (32 channels)
  int src = ei[e], dst = ei[E + e];
  float l = lat[e];
  float val = q[(long)dst * 128 + c] * (k[(long)src * 128 + c] + l * We[c]);
  for (int o = 16; o; o >>= 1) val += __shfl_down(val, o, 32);
  int h = c >> 5;
  if ((c & 31) == 0) {
    float a = val * 0.17677669529663687f;  // 1/sqrt(32)
    alpha[(long)e * 4 + h] = a;
    atomicMax(&amax[(long)dst * 4 + h], fkey(a));
  }
}

__global__ void edge_exp_kernel(const int* __restrict__ ei,
                                float* __restrict__ alpha,
                                const unsigned* __restrict__ amax,
                                float* __restrict__ den, int E) {
  long g = (long)blockIdx.x * blockDim.x + threadIdx.x;
  if (g >= (long)E * 4) return;
  int e = (int)(g >> 2), h = (int)(g & 3);
  int dst = ei[E + e];
  float ex = __expf(alpha[g] - fdec(amax[(long)dst * 4 + h]));
  alpha[g] = ex;  // reuse buffer as ex
  atomicAdd(&den[(long)dst * 4 + h], ex);
}

__global__ __launch_bounds__(128) void edge_scatter_kernel(
    const float* __restrict__ v, const float* __restrict__ We,
    const float* __restrict__ lat, const int* __restrict__ ei,
    const float* __restrict__ ex, const float* __restrict__ den,
    float* __restrict__ out, int E) {
  int e = blockIdx.x;
  int c = threadIdx.x;
  int h = c >> 5;
  int src = ei[e], dst = ei[E + e];
  float w = ex[(long)e * 4 + h] / (den[(long)dst * 4 + h] + 1e-16f);
  float val = (v[(long)src * 128 + c] + lat[e] * We[c]) * w;
  atomicAdd(&out[(long)dst * 128 + c], val);
}

// ---------------- gating: beta = sigmoid([out,xr,out-xr]@Wb) ----------------
__global__ __launch_bounds__(128) void gate_kernel(
    const float* __restrict__ out, const float* __restrict__ xr,
    const float* __restrict__ Wb, float* __restrict__ X, int N) {
  int n = blockIdx.x;
  int c = threadIdx.x;
  float o = out[(long)n * 128 + c], r = xr[(long)n * 128 + c];
  float p = o * Wb[c] + r * Wb[128 + c] + (o - r) * Wb[256 + c];
  for (int of = 16; of; of >>= 1) p += __shfl_down(p, of, 32);
  __shared__ float wsum[4];
  __shared__ float betaS;
  if ((c & 31) == 0) wsum[c >> 5] = p;
  __syncthreads();
  if (c == 0) {
    float s = wsum[0] + wsum[1] + wsum[2] + wsum[3];
    betaS = 1.f / (1.f + __expf(-s));
  }
  __syncthreads();
  float b = betaS;
  X[(long)n * 128 + c] = b * r + (1.f - b) * o;
}

// ---------------- batchnorm: column stats via LDS atomics -------------------
__global__ __launch_bounds__(256) void bn_stats_kernel(
    const float* __restrict__ Y, float* __restrict__ stats, int N, int M) {
  __shared__ float s1[32], s2[32];
  int t = threadIdx.x;
  if (t < M) { s1[t] = 0.f; s2[t] = 0.f; }
  __syncthreads();
  long total = (long)N * M;
  for (long g = (long)blockIdx.x * blockDim.x + t; g < total;
       g += (long)gridDim.x * blockDim.x) {
    float v = Y[g];
    int c = (int)(g % M);
    atomicAdd(&s1[c], v);  // ds_add_f32
    atomicAdd(&s2[c], v * v);
  }
  __syncthreads();
  if (t < M) {
    atomicAdd(&stats[t], s1[t]);
    atomicAdd(&stats[M + t], s2[t]);
  }
}

__global__ void bn_apply_kernel(const float* __restrict__ Y,
                                const float* __restrict__ stats,
                                const float* __restrict__ g,
                                const float* __restrict__ b,
                                float* __restrict__ X, int N, int M) {
  long id = (long)blockIdx.x * blockDim.x + threadIdx.x;
  if (id >= (long)N * M) return;
  int c = (int)(id % M);
  float inv = 1.f / (float)N;
  float mean = stats[c] * inv;
  float var = stats[M + c] * inv - mean * mean;
  X[id] = (Y[id] - mean) * rsqrtf(var + 1e-5f) * g[c] + b[c];
}

// ---------------- head ------------------------------------------------------
__global__ void logits1_kernel(const float* __restrict__ X,
                               const float* __restrict__ remW,
                               const float* __restrict__ remB,
                               const int* __restrict__ act,
                               float* __restrict__ out, int N) {
  int n = blockIdx.x * blockDim.x + threadIdx.x;
  if (n >= N) return;
  float s = remB[0];
  for (int i = 0; i < 64; ++i) s += X[(long)n * 64 + i] * remW[i];
  bool a = act[n] == 1;
  float maskf = a ? 0.f : NEGBIG;
  float rm = (n < 15) ? (a ? NEGBIG : 0.f) : maskf;
  out[n] = s + rm;
}

__global__ void argmax_gumbel_kernel(const float* __restrict__ logits,
                                     unsigned k0, unsigned k1,
                                     unsigned long long* slot, int N) {
  int n = blockIdx.x * blockDim.x + threadIdx.x;
  if (n >= N) return;
  float v = logits[n] + gumbel_of(k0, k1, (uint32_t)n);
  unsigned long long key = ((unsigned long long)fkey(v) << 32) |
                           (unsigned long long)(0xFFFFFFFFu - (unsigned)n);
  atomicMax(slot, key);  // ties -> smallest index wins
}

__global__ void rmax_kernel(const float* __restrict__ logits, unsigned* mx, int N) {
  int n = blockIdx.x * blockDim.x + threadIdx.x;
  if (n >= N) return;
  atomicMax(mx, fkey(logits[n]));
}

__global__ void sumexp_kernel(const float* __restrict__ logits,
                              const unsigned* __restrict__ mx,
                              float* sum, int N) {
  int n = blockIdx.x * blockDim.x + threadIdx.x;
  if (n >= N) return;
  atomicAdd(sum, __expf(logits[n] - fdec(*mx)));
}

__device__ inline int slot_idx(const unsigned long long* slot) {
  return (int)(0xFFFFFFFFu - (unsigned)(*slot & 0xFFFFFFFFull));
}

__global__ __launch_bounds__(64) void tvec_kernel(
    const float* __restrict__ X, const float* __restrict__ addW,
    const float* __restrict__ addB, const unsigned long long* slot,
    float* __restrict__ t) {
  int j = threadIdx.x;  // 64
  int a1 = slot_idx(slot);
  float s = addB[j];
  for (int i = 0; i < 64; ++i) s += X[(long)a1 * 64 + i] * addW[i * 64 + j];
  t[j] = tanhf(s);
}

__global__ void logits2_kernel(const float* __restrict__ X,
                               const float* __restrict__ t,
                               const int* __restrict__ act,
                               const unsigned long long* slot,
                               float* __restrict__ out, int N) {
  int n = blockIdx.x * blockDim.x + threadIdx.x;
  if (n >= N) return;
  int a1 = slot_idx(slot);
  float s = 0.f;
  for (int i = 0; i < 64; ++i) s += X[(long)n * 64 + i] * t[i];
  float maskf = (act[n] == 1) ? 0.f : NEGBIG;
  if (n == a1) maskf = 0.f;
  out[n] = s + maskf;
}

__global__ void finalize_kernel(const float* l1, const float* l2,
                                const unsigned long long* s1,
                                const unsigned long long* s2,
                                const unsigned* mx1, const float* sum1,
                                const unsigned* mx2, const float* sum2,
                                float* tail) {
  if (threadIdx.x == 0) {
    int a1 = slot_idx(s1), a2 = slot_idx(s2);
    tail[0] = (float)a1;
    tail[1] = (float)a2;
    tail[2] = l1[a1] - (fdec(*mx1) + __logf(*sum1));
    tail[3] = l2[a2] - (fdec(*mx2) + __logf(*sum2));
  }
}

// ---------------------------------------------------------------------------
extern "C" void kernel_launch(void* const* d_in, const int* in_sizes, int n_in,
                              void* d_out, int out_size, void* d_ws, size_t ws_size,
                              hipStream_t stream) {
  const int N = in_sizes[0];
  const int E = in_sizes[2];

  const float* node_type = (const float*)d_in[0];
  const float* requests  = (const float*)d_in[1];
  const float* latency   = (const float*)d_in[2];
  const int*   ei        = (const int*)d_in[3];
  const int*   act       = (const int*)d_in[4];
  const float* c0_Wq = (const float*)d_in[5],  *c0_bq = (const float*)d_in[6];
  const float* c0_Wk = (const float*)d_in[7],  *c0_bk = (const float*)d_in[8];
  const float* c0_Wv = (const float*)d_in[9],  *c0_bv = (const float*)d_in[10];
  const float* c0_We = (const float*)d_in[11];
  const float* c0_Ws = (const float*)d_in[12], *c0_bs = (const float*)d_in[13];
  const float* c0_Wb = (const float*)d_in[14];
  const float* t0_W  = (const float*)d_in[15], *t0_b  = (const float*)d_in[16];
  const float* bn0_g = (const float*)d_in[17], *bn0_b = (const float*)d_in[18];
  const float* cs_Wq = (const float*)d_in[19], *cs_bq = (const float*)d_in[20];
  const float* cs_Wk = (const float*)d_in[21], *cs_bk = (const float*)d_in[22];
  const float* cs_Wv = (const float*)d_in[23], *cs_bv = (const float*)d_in[24];
  const float* cs_We = (const float*)d_in[25];
  const float* cs_Ws = (const float*)d_in[26], *cs_bs = (const float*)d_in[27];
  const float* cs_Wb = (const float*)d_in[28];
  const float* ts_W  = (const float*)d_in[29], *ts_b  = (const float*)d_in[30];
  const float* bns_g = (const float*)d_in[31], *bns_b = (const float*)d_in[32];
  const float* emb_W1 = (const float*)d_in[33], *emb_b1 = (const float*)d_in[34];
  const float* emb_W2 = (const float*)d_in[35], *emb_b2 = (const float*)d_in[36];
  const float* emb_W3 = (const float*)d_in[37], *emb_b3 = (const float*)d_in[38];
  const float* rem_W  = (const float*)d_in[39], *rem_b  = (const float*)d_in[40];
  const float* add_W  = (const float*)d_in[41], *add_b  = (const float*)d_in[42];

  // ---- workspace carve-up ----
  char* wp = (char*)d_ws;
  auto alloc = [&](size_t bytes) -> void* {
    void* p = wp;
    wp += (bytes + 255) & ~(size_t)255;
    return p;
  };
  float* X    = (float*)alloc((size_t)N * 128 * 4);
  float* Q    = (float*)alloc((size_t)N * 128 * 4);
  float* Kb   = (float*)alloc((size_t)N * 128 * 4);
  float* V    = (float*)alloc((size_t)N * 128 * 4);
  float* XR   = (float*)alloc((size_t)N * 128 * 4);
  float* OUT  = (float*)alloc((size_t)N * 128 * 4);
  float* TBUF = (float*)alloc((size_t)N * 32 * 4);
  float* ALPHA = (float*)alloc((size_t)E * 4 * 4);
  unsigned* AMAX = (unsigned*)alloc((size_t)N * 4 * 4);
  float* DEN  = (float*)alloc((size_t)N * 4 * 4);
  float* X2   = (float*)alloc((size_t)N * 32 * 4);
  float* STATS = (float*)alloc(64 * 4);
  unsigned long long* SLOT1 = (unsigned long long*)alloc(8);
  unsigned long long* SLOT2 = (unsigned long long*)alloc(8);
  unsigned* MX1 = (unsigned*)alloc(4);
  unsigned* MX2 = (unsigned*)alloc(4);
  float* SUM1 = (float*)alloc(4);
  float* SUM2 = (float*)alloc(4);
  float* TVEC = (float*)alloc(64 * 4);

  float* logits1 = (float*)d_out;
  float* logits2 = (float*)d_out + N;
  float* tail    = (float*)d_out + 2 * N;

  const int rowTiles = (N + 15) / 16;
  const int gblocks = (rowTiles + 7) / 8;

  auto edge_phase = [&](const float* We, const float* Wb) {
    (void)hipMemsetAsync(AMAX, 0, (size_t)N * 4 * 4, stream);  // key 0 == floor
    (void)hipMemsetAsync(DEN, 0, (size_t)N * 4 * 4, stream);
    (void)hipMemsetAsync(OUT, 0, (size_t)N * 128 * 4, stream);
    edge_alpha_kernel<<<E, 128, 0, stream>>>(Q, Kb, We, latency, ei, ALPHA, AMAX, E);
    edge_exp_kernel<<<(int)(((long)E * 4 + 255) / 256), 256, 0, stream>>>(
        ei, ALPHA, AMAX, DEN, E);
    edge_scatter_kernel<<<E, 128, 0, stream>>>(V, We, latency, ei, ALPHA, DEN, OUT, E);
    gate_kernel<<<N, 128, 0, stream>>>(OUT, XR, Wb, X, N);
  };

  auto tbn = [&](const float* tW, const float* tb, const float* g,
                 const float* b) {
    gemm_wmma<128, 32><<<gblocks, 256, 0, stream>>>(X, tW, tb, TBUF, N, 1);
    (void)hipMemsetAsync(STATS, 0, 64 * 4, stream);
    bn_stats_kernel<<<512, 256, 0, stream>>>(TBUF, STATS, N, 32);
    bn_apply_kernel<<<(int)(((long)N * 32 + 255) / 256), 256, 0, stream>>>(
        TBUF, STATS, g, b, X, N, 32);
  };

  // ---- layer 0: x = stack(node_type, requests) ; K=2 outer-product GEMMs ---
  stack_kernel<<<(N + 255) / 256, 256, 0, stream>>>(node_type, requests, X2, N);
  {
    int eb = (int)(((long)N * 128 + 255) / 256);
    gemm_k2_kernel<<<eb, 256, 0, stream>>>(X2, c0_Wq, c0_bq, Q, N);
    gemm_k2_kernel<<<eb, 256, 0, stream>>>(X2, c0_Wk, c0_bk, Kb, N);
    gemm_k2_kernel<<<eb, 256, 0, stream>>>(X2, c0_Wv, c0_bv, V, N);
    gemm_k2_kernel<<<eb, 256, 0, stream>>>(X2, c0_Ws, c0_bs, XR, N);
  }
  edge_phase(c0_We, c0_Wb);
  tbn(t0_W, t0_b, bn0_g, bn0_b);

  // ---- layers 1..4: K=32 WMMA GEMMs ----
  for (int i = 0; i < 4; ++i) {
    const size_t wO = (size_t)i * 32 * 128, bO = (size_t)i * 128;
    (void)hipMemcpyAsync(X2, X, (size_t)N * 32 * 4, hipMemcpyDeviceToDevice, stream);
    gemm_wmma<32, 128><<<gblocks, 256, 0, stream>>>(X2, cs_Wq + wO, cs_bq + bO, Q,  N, 0);
    gemm_wmma<32, 128><<<gblocks, 256, 0, stream>>>(X2, cs_Wk + wO, cs_bk + bO, Kb, N, 0);
    gemm_wmma<32, 128><<<gblocks, 256, 0, stream>>>(X2, cs_Wv + wO, cs_bv + bO, V,  N, 0);
    gemm_wmma<32, 128><<<gblocks, 256, 0, stream>>>(X2, cs_Ws + wO, cs_bs + bO, XR, N, 0);
    edge_phase(cs_We + (size_t)i * 128, cs_Wb + (size_t)i * 384);
    tbn(ts_W + (size_t)i * 128 * 32, ts_b + (size_t)i * 32,
        bns_g + (size_t)i * 32, bns_b + (size_t)i * 32);
  }

  // ---- final MLP: 32 -> 128 -> 64 -> 64 ----
  gemm_wmma<32, 128><<<gblocks, 256, 0, stream>>>(X, emb_W1, emb_b1, Q, N, 1);
  gemm_wmma<128, 64><<<gblocks, 256, 0, stream>>>(Q, emb_W2, emb_b2, Kb, N, 1);
  gemm_wmma<64, 64><<<gblocks, 256, 0, stream>>>(Kb, emb_W3, emb_b3, V, N, 0);
  const float* FX = V;  // N x 64 final features

  // ---- head: gumbel argmax x2 + log-softmax ----
  uint32_t k1a, k1b, k2a, k2b;
  threefry2x32(0u, 42u, 0u, 1u, k1a, k1b);
  threefry2x32(0u, 42u, 2u, 3u, k2a, k2b);

  (void)hipMemsetAsync(SLOT1, 0, 8, stream);
  (void)hipMemsetAsync(SLOT2, 0, 8, stream);
  (void)hipMemsetAsync(MX1, 0, 4, stream);
  (void)hipMemsetAsync(MX2, 0, 4, stream);
  (void)hipMemsetAsync(SUM1, 0, 4, stream);
  (void)hipMemsetAsync(SUM2, 0, 4, stream);

  int nb = (N + 255) / 256;
  logits1_kernel<<<nb, 256, 0, stream>>>(FX, rem_W, rem_b, act, logits1, N);
  rmax_kernel<<<nb, 256, 0, stream>>>(logits1, MX1, N);
  argmax_gumbel_kernel<<<nb, 256, 0, stream>>>(logits1, k1a, k1b, SLOT1, N);
  sumexp_kernel<<<nb, 256, 0, stream>>>(logits1, MX1, SUM1, N);

  tvec_kernel<<<1, 64, 0, stream>>>(FX, add_W, add_b, SLOT1, TVEC);
  logits2_kernel<<<nb, 256, 0, stream>>>(FX, TVEC, act, SLOT1, logits2, N);
  rmax_kernel<<<nb, 256, 0, stream>>>(logits2, MX2, N);
  argmax_gumbel_kernel<<<nb, 256, 0, stream>>>(logits2, k2a, k2b, SLOT2, N);
  sumexp_kernel<<<nb, 256, 0, stream>>>(logits2, MX2, SUM2, N);

  finalize_kernel<<<1, 64, 0, stream>>>(logits1, logits2, SLOT1, SLOT2,
                                        MX1, SUM1, MX2, SUM2, tail);
  (void)n_in; (void)out_size; (void)ws_size;
}